// Embeder_2800318677560
// MI455X (gfx1250) — compile-verified
//
#include <hip/hip_runtime.h>
#include <cstdint>
#include <cstddef>

// ---------------------------------------------------------------------------
// out[b,t,:] = weight[x[b,t],:]
//            + (x[b,t] not an end symbol && exists end symbol at t' > t)
//              ? 1.5f * weight[ x[b, min{t'>t : x[b,t'] in end_ids}], : ] : 0
//
// Roofline: 512 MB f32 stores dominate (weight table = 512 KB, L2-resident;
// index traffic ~1 MB). ~22 us floor at 23.3 TB/s. Pure data movement -> no
// WMMA; CDNA5 features used: async global->LDS staging (ASYNCcnt),
// s_wait_asynccnt, wave32 LDS suffix scan, b128 non-temporal stores.
// ---------------------------------------------------------------------------

#define TSZ   1024   // T
#define DSZ   512    // D
#define BLOCK 1024   // 32 wave32s, one block per batch row

typedef float f4 __attribute__((ext_vector_type(4)));

// Async global -> LDS copy, 4 bytes per lane. Tracked by ASYNCcnt.
// VDST operand = per-lane LDS byte offset, VADDR = per-lane 64-bit address.
__device__ __forceinline__ void async_ld_b32(void* lds_dst, const void* gsrc) {
  uint32_t loff = (uint32_t)(uintptr_t)lds_dst;  // low 32 bits = LDS offset
  asm volatile("global_load_async_to_lds_b32 %0, %1, off"
               :: "v"(loff), "v"(gsrc)
               : "memory");
}

__device__ __forceinline__ void wait_async0() {
#if __has_builtin(__builtin_amdgcn_s_wait_asynccnt)
  __builtin_amdgcn_s_wait_asynccnt(0);
#else
  asm volatile("s_wait_asynccnt 0x0" ::: "memory");
#endif
}

template <int E>
__global__ __launch_bounds__(BLOCK)
void embed_suffix_kernel(const float* __restrict__ weight,
                         const int*   __restrict__ x,
                         const int*   __restrict__ end_ids,
                         float* __restrict__ out)
{
  __shared__ int   tok[TSZ];    // token ids for this batch row      (4 KB)
  __shared__ int   sfx[TSZ];    // suffix-scan workspace             (4 KB)
  __shared__ short code[TSZ];   // addend weight-row per t, or -1    (2 KB)

  const int b   = blockIdx.x;
  const int tid = threadIdx.x;

  // ---- Phase 1: stage this row's 1024 token ids into LDS (async DMA) ----
  async_ld_b32(&tok[tid], x + (size_t)b * TSZ + tid);

  // end-symbol ids: uniform, compile-time-unrolled scalar loads; overlaps DMA
  int eid[E];
#pragma unroll
  for (int e = 0; e < E; ++e) eid[e] = end_ids[e];

  wait_async0();        // drain this wave's ASYNCcnt before the barrier
  __syncthreads();

  // ---- Phase 2: suffix scan "row id of first end symbol at t' >= t" ----
  const int v = tok[tid];
  bool is_end = false;
#pragma unroll
  for (int e = 0; e < E; ++e) is_end = is_end || (v == eid[e]);
  // addend row for a matched end symbol is the token value itself:
  // end_vecs[idx] = 1.5 * weight[end_ids[idx]] and end_ids[idx] == v.
  sfx[tid] = is_end ? v : -1;
  __syncthreads();

#pragma unroll
  for (int off = 1; off < TSZ; off <<= 1) {
    const int mine  = sfx[tid];
    const int other = (tid + off < TSZ) ? sfx[tid + off] : -1;
    __syncthreads();
    sfx[tid] = (mine >= 0) ? mine : other;
    __syncthreads();
  }

  // nearest end symbol strictly after t; suppress add at end positions
  const int cur = (tid + 1 < TSZ) ? sfx[tid + 1] : -1;
  code[tid] = (short)(is_end ? -1 : cur);
  __syncthreads();

  // ---- Phase 3: stream the output (store-bandwidth bound) ----
  // Lanes (tid & 127) own a fixed 16-byte d-slice; tid>>7 selects the
  // starting t. 128 lanes share one t -> LDS broadcast reads, uniform
  // branch on the addend, coalesced b128 loads + NT b128 stores.
  // Unroll x4 for memory-level parallelism against the HBM store floor.
  const int d0 = (tid & 127) << 2;
  float* __restrict__ orow = out + (size_t)b * TSZ * DSZ;

#pragma unroll 4
  for (int t = (tid >> 7); t < TSZ; t += 8) {
    const int r = tok[t];
    const int c = code[t];
    f4 w = *(const f4*)(weight + (size_t)r * DSZ + d0);
    if (c >= 0) {
      const f4 a = *(const f4*)(weight + (size_t)c * DSZ + d0);
      w.x = fmaf(1.5f, a.x, w.x);
      w.y = fmaf(1.5f, a.y, w.y);
      w.z = fmaf(1.5f, a.z, w.z);
      w.w = fmaf(1.5f, a.w, w.w);
    }
    // output is write-once, never re-read: non-temporal store
    __builtin_nontemporal_store(w, (f4*)(orow + (size_t)t * DSZ + d0));
  }
}

// Dynamic-E fallback (same algorithm, runtime loop over end_ids).
__global__ __launch_bounds__(BLOCK)
void embed_suffix_kernel_dyn(const float* __restrict__ weight,
                             const int*   __restrict__ x,
                             const int*   __restrict__ end_ids,
                             int n_end,
                             float* __restrict__ out)
{
  __shared__ int   tok[TSZ];
  __shared__ int   sfx[TSZ];
  __shared__ short code[TSZ];

  const int b   = blockIdx.x;
  const int tid = threadIdx.x;

  async_ld_b32(&tok[tid], x + (size_t)b * TSZ + tid);
  wait_async0();
  __syncthreads();

  const int v = tok[tid];
  bool is_end = false;
  for (int e = 0; e < n_end; ++e) is_end = is_end || (v == end_ids[e]);
  sfx[tid] = is_end ? v : -1;
  __syncthreads();

  for (int off = 1; off < TSZ; off <<= 1) {
    const int mine  = sfx[tid];
    const int other = (tid + off < TSZ) ? sfx[tid + off] : -1;
    __syncthreads();
    sfx[tid] = (mine >= 0) ? mine : other;
    __syncthreads();
  }

  const int cur = (tid + 1 < TSZ) ? sfx[tid + 1] : -1;
  code[tid] = (short)(is_end ? -1 : cur);
  __syncthreads();

  const int d0 = (tid & 127) << 2;
  float* __restrict__ orow = out + (size_t)b * TSZ * DSZ;

#pragma unroll 4
  for (int t = (tid >> 7); t < TSZ; t += 8) {
    const int r = tok[t];
    const int c = code[t];
    f4 w = *(const f4*)(weight + (size_t)r * DSZ + d0);
    if (c >= 0) {
      const f4 a = *(const f4*)(weight + (size_t)c * DSZ + d0);
      w.x = fmaf(1.5f, a.x, w.x);
      w.y = fmaf(1.5f, a.y, w.y);
      w.z = fmaf(1.5f, a.z, w.z);
      w.w = fmaf(1.5f, a.w, w.w);
    }
    __builtin_nontemporal_store(w, (f4*)(orow + (size_t)t * DSZ + d0));
  }
}

extern "C" void kernel_launch(void* const* d_in, const int* in_sizes, int n_in,
                              void* d_out, int out_size, void* d_ws, size_t ws_size,
                              hipStream_t stream) {
  const float* weight  = (const float*)d_in[0];   // [256, 512] f32
  const int*   x       = (const int*)d_in[1];     // [256, 1024] int
  const int*   end_ids = (const int*)d_in[2];     // [2] int
  const int    n_end   = in_sizes[2];
  const int    B       = in_sizes[1] / TSZ;
  float*       out     = (float*)d_out;

  switch (n_end) {
    case 1:
      embed_suffix_kernel<1><<<B, BLOCK, 0, stream>>>(weight, x, end_ids, out);
      break;
    case 2:
      embed_suffix_kernel<2><<<B, BLOCK, 0, stream>>>(weight, x, end_ids, out);
      break;
    case 3:
      embed_suffix_kernel<3><<<B, BLOCK, 0, stream>>>(weight, x, end_ids, out);
      break;
    case 4:
      embed_suffix_kernel<4><<<B, BLOCK, 0, stream>>>(weight, x, end_ids, out);
      break;
    default:
      embed_suffix_kernel_dyn<<<B, BLOCK, 0, stream>>>(weight, x, end_ids,
                                                       n_end, out);
      break;
  }
}